// LocallyConnected2d_77068893159597
// MI455X (gfx1250) — compile-verified
//
#include <hip/hip_runtime.h>

typedef __attribute__((ext_vector_type(2))) float v2f;
typedef __attribute__((ext_vector_type(8))) float v8f;

#define BATCH   16
#define IN_C    32
#define IMG     64
#define IN_FEAT 288          // 32 * 3 * 3
#define OUT_CH  64
#define L_TOTAL 4096
#define PSTRIDE 292          // padded LDS row stride (floats): 292%64=36 -> conflict-free b64 reads

__global__ __launch_bounds__(128)
void lc2d_wmma_kernel(const float* __restrict__ x,
                      const float* __restrict__ weight,
                      const float* __restrict__ bias,
                      float* __restrict__ out) {
    __shared__ float patches[BATCH * PSTRIDE];   // 16 x 288 (padded) ~ 18.7 KB

    const int l    = blockIdx.x;       // spatial location 0..4095
    const int oh   = l >> 6;
    const int ow   = l & 63;
    const int tid  = threadIdx.x;      // 0..127
    const int lane = tid & 31;
    const int wave = tid >> 5;         // 0..3 -> N tile of 16 output channels

    // ---- stage unfold(x) patch matrix (16 x 288) into LDS ----
    for (int idx = tid; idx < BATCH * IN_FEAT; idx += 128) {
        const int b  = idx / IN_FEAT;
        const int f  = idx - b * IN_FEAT;       // c*9 + ki*3 + kj (torch unfold order)
        const int c  = f / 9;
        const int r  = f - c * 9;
        const int ki = r / 3;
        const int kj = r - ki * 3;
        const int ih = oh - 1 + ki;
        const int iw = ow - 1 + kj;
        float v = 0.0f;
        if ((unsigned)ih < (unsigned)IMG && (unsigned)iw < (unsigned)IMG)
            v = x[(((size_t)b * IN_C + c) * IMG + ih) * IMG + iw];
        patches[b * PSTRIDE + f] = v;
    }
    __syncthreads();

    // ---- per-wave 16(batch) x 16(out_ch) tile, K = 288 via V_WMMA_F32_16X16X4_F32 ----
    const int n0 = wave * 16;
    const int nl = lane & 15;          // N (and M for A-frag) sub-index within tile
    const int kh = lane >> 4;          // selects K pair {2*kh, 2*kh+1} within each k-step

    // B fragment source: weight[l][k][n0+nl], k stride = OUT_CH
    const float* wl = weight + (size_t)l * (IN_FEAT * OUT_CH) + (n0 + nl);
    // A fragment source: patches[m = nl][k], per ISA 16x4 f32 A layout
    const float* ap = &patches[nl * PSTRIDE];

    // accumulator initialized from bias (depends only on out_ch = N)
    const float bv = bias[(size_t)l * OUT_CH + n0 + nl];
    v8f acc = (v8f){bv, bv, bv, bv, bv, bv, bv, bv};

    #pragma unroll 8
    for (int k0 = 0; k0 < IN_FEAT; k0 += 4) {
        const int kb = k0 + 2 * kh;
        // A: lane holds A[m = nl][kb], A[m][kb+1]  (8B-aligned LDS load)
        v2f a = *(const v2f*)(ap + kb);
        // B: lane holds B[kb][n], B[kb+1][n] ; single-use weight stream -> non-temporal
        v2f b;
        b.x = __builtin_nontemporal_load(wl + (size_t)kb * OUT_CH);
        b.y = __builtin_nontemporal_load(wl + (size_t)(kb + 1) * OUT_CH);
        // 8 args: (neg_a, A, neg_b, B, c_mod, C, reuse_a, reuse_b)
        acc = __builtin_amdgcn_wmma_f32_16x16x4_f32(
            false, a, false, b, (short)0, acc, false, false);
    }

    // ---- store D: VGPR i -> M = i + 8*kh (batch), N = n0+nl (out_ch) ----
    float* op = out + (size_t)(n0 + nl) * L_TOTAL + l;   // + b * OUT_CH*L below
    #pragma unroll
    for (int i = 0; i < 8; ++i) {
        const int m = i + 8 * kh;                        // batch index
        op[(size_t)m * (OUT_CH * L_TOTAL)] = acc[i];
    }
}

extern "C" void kernel_launch(void* const* d_in, const int* in_sizes, int n_in,
                              void* d_out, int out_size, void* d_ws, size_t ws_size,
                              hipStream_t stream) {
    const float* x      = (const float*)d_in[0];   // (16, 32, 64, 64)
    const float* weight = (const float*)d_in[1];   // (4096, 288, 64)
    const float* bias   = (const float*)d_in[2];   // (4096, 64)
    float*       out    = (float*)d_out;           // (16, 64, 64, 64)

    dim3 grid(L_TOTAL);
    dim3 block(128);
    hipLaunchKernelGGL(lc2d_wmma_kernel, grid, block, 0, stream,
                       x, weight, bias, out);
}